// SLMUSEEmbeddings_23940147708280
// MI455X (gfx1250) — compile-verified
//
#include <hip/hip_runtime.h>
#include <hip/hip_bf16.h>

#define NB 16      // B
#define NS 16      // S
#define NL 128     // L
#define NA 8       // A
#define NT 8       // T
#define ND 768     // D
#define WSTRIDE 132  // padded LDS row stride for W (bank-conflict-free WMMA A reads)

#define SENT_SZ (NB * NS * ND)        // 196608 floats
#define ARG_SZ  (NB * NS * NA * ND)   // 1572864 floats

typedef __attribute__((ext_vector_type(2))) float v2f;
typedef __attribute__((ext_vector_type(8))) float v8f;

// Physical W row -> logical output row (0=sent, 1..8 pred, 9..16 arg0,
// 17..24 arg1), or -1 for zero-padding rows. Permuted so that in the second
// D tile every D-VGPR pairs two valid rows (r=0..3) or two dead rows (r=5..7),
// with a single half-wave-guarded row at r=4 -> branchless stores.
__device__ __forceinline__ int phys_to_logical(int p) {
    if (p < 20) return p;              // 0..19  -> logical 0..19
    if (p == 20) return 24;            // 20     -> logical 24
    if (p >= 24 && p <= 27) return p - 4; // 24..27 -> logical 20..23
    return -1;                         // 21..23, 28..31 dead
}

// One workgroup per (b,s). 256 threads = 8 waves (wave32).
// Phase 1: build scaled weight matrix W[32 x 128] in LDS (division folded in),
//          plus per-physical-row 32-bit output offsets from the d_out base.
// Phase 2: each wave computes both 16-row M-tiles for 6 N-tiles:
//          D[32 x 768] = W x E via v_wmma_f32_16x16x4_f32, B shared between
//          the two accumulators; global (not flat) non-temporal stores.
__global__ __launch_bounds__(256)
void slmuse_embed_kernel(const int* __restrict__ sid,
                         const int* __restrict__ amask,
                         const int* __restrict__ pid,
                         const int* __restrict__ a0id,
                         const int* __restrict__ a1id,
                         const float* __restrict__ emb,
                         float* __restrict__ out_base) {
    __shared__ float    Wm[32 * WSTRIDE];
    __shared__ int      s_sid[NL];
    __shared__ int      s_mask[NL];
    __shared__ int      s_ids[3 * NA * NT];   // [pred | arg0 | arg1]
    __shared__ float    s_inv[32];
    __shared__ unsigned s_off[32];            // element offset into d_out per physical row

    const int bs  = blockIdx.x;    // 0..255  (b*16+s)
    const int tid = threadIdx.x;   // 0..255

    // ---- stage ids + build destination offset table ----
    if (tid < NL) {
        s_sid[tid]  = sid[bs * NL + tid];
        s_mask[tid] = amask[bs * NL + tid];
    }
    if (tid < NA * NT) {
        s_ids[tid]               = pid[bs * NA * NT + tid];
        s_ids[NA * NT + tid]     = a0id[bs * NA * NT + tid];
        s_ids[2 * NA * NT + tid] = a1id[bs * NA * NT + tid];
    }
    if (tid < 32) {
        const int l = phys_to_logical(tid);
        unsigned off = 0;   // dead rows: never stored
        if (l == 0)       off = (unsigned)(bs * ND);
        else if (l <= 8)  off = (unsigned)(SENT_SZ + (bs * NA + (l - 1)) * ND);
        else if (l <= 16) off = (unsigned)(SENT_SZ + ARG_SZ + (bs * NA + (l - 9)) * ND);
        else if (l <= 24) off = (unsigned)(SENT_SZ + 2 * ARG_SZ + (bs * NA + (l - 17)) * ND);
        s_off[tid] = off;
    }
    __syncthreads();

    // ---- raw W: 32 physical rows x 128 cols, 16 entries per thread ----
    for (int e = tid; e < 32 * NL; e += 256) {
        const int p = e >> 7;       // physical row
        const int c = e & (NL - 1); // col
        const int l = phys_to_logical(p);
        float w;
        if (l == 0) {
            w = (float)s_mask[c];
        } else if (l > 0) {
            const int base = (l - 1) * NT;   // logical rows 1..24 -> 24 arg slots
            const int sv = s_sid[c];
            int cnt = 0;
#pragma unroll
            for (int t = 0; t < NT; ++t) {
                const int id = s_ids[base + t];
                cnt += (id != 0 && id == sv) ? 1 : 0;
            }
            w = (float)cnt;
        } else {
            w = 0.0f;                        // dead padding row
        }
        Wm[p * WSTRIDE + c] = w;
    }
    __syncthreads();

    // ---- per-row counts -> inverse scale (fold the division into W) ----
    if (tid < 32) {
        float c = 0.0f;
        for (int x = 0; x < NL; ++x) c += Wm[tid * WSTRIDE + x];
        float inv;
        if (tid == 0)
            inv = 1.0f / fmaxf(c, 1.0f);            // mean-pool: clamp(min=1)
        else
            inv = (c > 0.0f) ? (1.0f / c) : 0.0f;   // zero rows stay zero
        s_inv[tid] = inv;
    }
    __syncthreads();
    for (int e = tid; e < 32 * NL; e += 256) {
        const int p = e >> 7;
        const int c = e & (NL - 1);
        Wm[p * WSTRIDE + c] *= s_inv[p];
    }
    __syncthreads();

    // ---- GEMM: D = W x E via V_WMMA_F32_16X16X4_F32, both M tiles per wave ----
    const int lane = tid & 31;
    const int wave = tid >> 5;          // 0..7
    const int h    = lane >> 4;         // half-wave (K split / D row split)
    const int ln   = lane & 15;
    const int ntBase = wave * 6;        // 48 N-tiles / 8 waves

    const float* E = emb + (size_t)bs * NL * ND;
    const float* Wrow0 = &Wm[ln * WSTRIDE];        // A rows 0..15
    const float* Wrow1 = &Wm[(16 + ln) * WSTRIDE]; // A rows 16..31

    // hoist per-lane destination pointers (constant across N-tiles)
    float* d0[8];
    float* d1[5];
#pragma unroll
    for (int r = 0; r < 8; ++r) d0[r] = out_base + s_off[r + 8 * h];      // rows 0..15
#pragma unroll
    for (int r = 0; r < 5; ++r) d1[r] = out_base + s_off[16 + r + 8 * h]; // tile-1 rows

    for (int j = 0; j < 6; ++j) {
        const int n = (ntBase + j) * 16 + ln;   // B/D column for this lane
        v8f acc0 = {};
        v8f acc1 = {};
#pragma unroll 4
        for (int k = 0; k < NL; k += 4) {
            v2f a0, a1, b;
            // A 16x4: VGPR0 = K{2h}, VGPR1 = K{2h+1}
            a0.x = Wrow0[k + 2 * h];
            a0.y = Wrow0[k + 2 * h + 1];
            a1.x = Wrow1[k + 2 * h];
            a1.y = Wrow1[k + 2 * h + 1];
            // B 4x16: shared by both M tiles (64B contiguous per half-wave)
            b.x = E[(size_t)(k + 2 * h) * ND + n];
            b.y = E[(size_t)(k + 2 * h + 1) * ND + n];
            acc0 = __builtin_amdgcn_wmma_f32_16x16x4_f32(
                false, a0, false, b, (short)0, acc0, false, false);
            acc1 = __builtin_amdgcn_wmma_f32_16x16x4_f32(
                false, a1, false, b, (short)0, acc1, false, false);
        }
        // Tile 0 (physical rows 0..15): all valid, unconditional.
#pragma unroll
        for (int r = 0; r < 8; ++r)
            __builtin_nontemporal_store(acc0[r], d0[r] + n);
        // Tile 1: r=0..3 valid for both halves; r=4 valid only for h==0;
        //         r=5..7 dead (no store).
#pragma unroll
        for (int r = 0; r < 4; ++r)
            __builtin_nontemporal_store(acc1[r], d1[r] + n);
        if (h == 0)
            __builtin_nontemporal_store(acc1[4], d1[4] + n);
    }
}

extern "C" void kernel_launch(void* const* d_in, const int* in_sizes, int n_in,
                              void* d_out, int out_size, void* d_ws, size_t ws_size,
                              hipStream_t stream) {
    (void)in_sizes; (void)n_in; (void)out_size; (void)d_ws; (void)ws_size;

    const int*   sid   = (const int*)d_in[0];   // [16,16,128]
    const int*   amask = (const int*)d_in[1];   // [16,16,128]
    const int*   pid   = (const int*)d_in[2];   // [16,16,8,8]
    const int*   a0id  = (const int*)d_in[3];   // [16,16,8,8]
    const int*   a1id  = (const int*)d_in[4];   // [16,16,8,8]
    const float* emb   = (const float*)d_in[5]; // [16,16,128,768]

    dim3 grid(NB * NS);   // 256 workgroups, one per (b,s)
    dim3 block(256);      // 8 wave32 waves
    slmuse_embed_kernel<<<grid, block, 0, stream>>>(
        sid, amask, pid, a0id, a1id, emb, (float*)d_out);
}